// DGCNN_82300163326097
// MI455X (gfx1250) — compile-verified
//
#include <hip/hip_runtime.h>
#include <hip/hip_bf16.h>
#include <math.h>

typedef __attribute__((ext_vector_type(2))) float v2f;
typedef __attribute__((ext_vector_type(8))) float v8f;

#define NPTS 4096
#define NB   4
#define KNN  20

// ---------------------------------------------------------------------------
// Squared norms: xx[b][n] = sum_c x[b][c][n]^2
// ---------------------------------------------------------------------------
__global__ void xx_kernel(const float* __restrict__ xin, int bstride, int C,
                          float* __restrict__ xx) {
    int t = blockIdx.x * blockDim.x + threadIdx.x;
    if (t >= NB * NPTS) return;
    int b = t / NPTS, n = t - b * NPTS;
    const float* xb = xin + (long)b * bstride;
    float s = 0.f;
    for (int c = 0; c < C; ++c) {
        float v = xb[c * NPTS + n];
        s += v * v;
    }
    xx[t] = s;
}

// ---------------------------------------------------------------------------
// Fused kNN. Block = 4 waves, each wave owns 64 of the 256 column tiles for
// one 16-row tile. A-fragments hoisted into registers (invariant over column
// tiles). Per-wave top-20 lists in LDS, merged at the end (lists sorted
// descending -> early-exit merge).
// ---------------------------------------------------------------------------
template <int C>
__global__ __launch_bounds__(128) void knn_kernel(const float* __restrict__ xin,
                                                  int bstride,
                                                  const float* __restrict__ xx,
                                                  int* __restrict__ idx_out) {
    __shared__ float tile[4][16][16];
    __shared__ float lv[4][16][KNN];
    __shared__ int   li[4][16][KNN];

    int wid = blockIdx.x;            // b * (N/16) + rowtile
    int b   = wid >> 8;              // N/16 = 256
    int rt  = wid & 255;
    int r0  = rt * 16;
    int wv   = threadIdx.x >> 5;
    int lane = threadIdx.x & 31;
    int hi = lane >> 4, lm = lane & 15;

    const float* xb  = xin + (long)b * bstride;
    const float* xxb = xx + b * NPTS;

    if (lane < 16) {
        for (int j = 0; j < KNN; ++j) { lv[wv][lane][j] = -3.0e38f; li[wv][lane][j] = 0; }
    }

    // Hoist A fragments (16 rows x C) into registers.
    v2f afrag[C / 4];
#pragma unroll
    for (int cc = 0; cc < C / 4; ++cc) {
        afrag[cc].x = xb[(4 * cc + 2 * hi) * NPTS + r0 + lm];
        afrag[cc].y = xb[(4 * cc + 2 * hi + 1) * NPTS + r0 + lm];
    }
    float xr[8];
#pragma unroll
    for (int v = 0; v < 8; ++v) xr[v] = xxb[r0 + v + 8 * hi];

    __syncthreads();

    for (int it = 0; it < 64; ++it) {
        int ct = wv * 64 + it;
        int c0 = ct * 16;
        // prefetch next column slab (global_prefetch_b8)
        if (it + 1 < 64)
            __builtin_prefetch(&xb[(2 * hi) * NPTS + c0 + 16 + lm], 0, 1);
        v8f acc = {};
#pragma unroll
        for (int cc = 0; cc < C / 4; ++cc) {
            v2f bm;
            bm.x = xb[(4 * cc + 2 * hi) * NPTS + c0 + lm];
            bm.y = xb[(4 * cc + 2 * hi + 1) * NPTS + c0 + lm];
            acc = __builtin_amdgcn_wmma_f32_16x16x4_f32(false, afrag[cc], false, bm,
                                                        (short)0, acc, false, false);
        }
        float xn = xxb[c0 + lm];
#pragma unroll
        for (int v = 0; v < 8; ++v)
            tile[wv][v + 8 * hi][lm] = 2.0f * acc[v] - xr[v] - xn;
        __syncthreads();
        if (lane < 16) {
            float* Lv = &lv[wv][lane][0];
            int*   Li = &li[wv][lane][0];
            for (int j = 0; j < 16; ++j) {
                float v = tile[wv][lane][j];
                if (v > Lv[KNN - 1]) {
                    int p = KNN - 1;
                    while (p > 0 && Lv[p - 1] < v) {
                        Lv[p] = Lv[p - 1]; Li[p] = Li[p - 1]; --p;
                    }
                    Lv[p] = v; Li[p] = c0 + j;
                }
            }
        }
        __syncthreads();
    }

    // Merge the 4 per-wave lists into lv[0] (each sorted descending).
    if (threadIdx.x < 16) {
        int row = threadIdx.x;
        float* Lv = &lv[0][row][0];
        int*   Li = &li[0][row][0];
        for (int wvi = 1; wvi < 4; ++wvi) {
            for (int j = 0; j < KNN; ++j) {
                float v = lv[wvi][row][j];
                if (v <= Lv[KNN - 1]) break;   // rest are smaller
                int p = KNN - 1;
                while (p > 0 && Lv[p - 1] < v) {
                    Lv[p] = Lv[p - 1]; Li[p] = Li[p - 1]; --p;
                }
                Lv[p] = v; Li[p] = li[wvi][row][j];
            }
        }
        int* dst = idx_out + ((long)b * NPTS + r0 + row) * KNN;
        for (int j = 0; j < KNN; ++j) dst[j] = Li[j];
    }
}

// ---------------------------------------------------------------------------
__global__ void zero_kernel(float* p, int n) {
    int i = blockIdx.x * blockDim.x + threadIdx.x;
    if (i < n) p[i] = 0.f;
}

// ---------------------------------------------------------------------------
// Build edge-feature tile (16 rows x 2C) in LDS.
//  - diff half (neigh - xt): VALU path (gather + subtract), row-fastest so the
//    subtrahend loads are broadcast-like across lanes.
//  - xt half: pure copy -> async global->LDS DMA (ASYNCcnt), overlapped with
//    the VALU half, fenced with s_wait_asynccnt before the barrier.
// ---------------------------------------------------------------------------
template <int C>
__device__ __forceinline__ void build_feat(float (*feat)[2 * C + 2],
                                           const float* __restrict__ xb,
                                           const int* __restrict__ idxb,
                                           int base, int t, int lane) {
    // xt half: async DMA, issue first so it overlaps the gather/subtract.
#pragma unroll 4
    for (int e = lane; e < 16 * C; e += 32) {
        int rr = e & 15, c = e >> 4;
        int mf = base + t * 16 + rr;
        int n = mf / KNN;
        unsigned ldsa = (unsigned)(unsigned long long)&feat[rr][C + c];
        unsigned long long ga = (unsigned long long)&xb[c * NPTS + n];
        asm volatile("global_load_async_to_lds_b32 %0, %1, off"
                     :: "v"(ldsa), "v"(ga) : "memory");
    }
    // diff half (VALU)
#pragma unroll 4
    for (int e = lane; e < 16 * C; e += 32) {
        int rr = e & 15, c = e >> 4;
        int mf = base + t * 16 + rr;
        int n = mf / KNN, kk = mf - n * KNN;
        int j = idxb[n * KNN + kk];
        feat[rr][c] = xb[c * NPTS + j] - xb[c * NPTS + n];
    }
    asm volatile("s_wait_asynccnt 0" ::: "memory");
}

// ---------------------------------------------------------------------------
// EdgeConv pass 1: GEMM (feat[327680 x 2C] * w^T[2C x F]) via WMMA;
// accumulate per-channel sum / sumsq (BN over B,N,k) in LDS, flush atomically.
// One wave per (b, group-of-4 points) = 80 rows = 5 M-tiles.
// ---------------------------------------------------------------------------
template <int C, int F>
__global__ __launch_bounds__(32) void edge_stats_kernel(
    const float* __restrict__ xin, int bstride,
    const float* __restrict__ w, const int* __restrict__ idx,
    float* __restrict__ gsum, float* __restrict__ gsq) {
    constexpr int TC = 2 * C;
    constexpr int FP = TC + 2;               // row pad: avoid LDS bank conflicts
    __shared__ float feat[16][FP];
    __shared__ float lsum[F], lsq[F];

    int lane = threadIdx.x;
    int hi = lane >> 4, lm = lane & 15;
    int wid = blockIdx.x;            // b * (N/4) + ngrp
    int b = wid >> 10, ngrp = wid & 1023;
    int base = ngrp * 80;            // flat row base (n*KNN + kk) within batch

    for (int e = lane; e < F; e += 32) { lsum[e] = 0.f; lsq[e] = 0.f; }
    const float* xb   = xin + (long)b * bstride;
    const int*   idxb = idx + (long)b * NPTS * KNN;

    for (int t = 0; t < 5; ++t) {
        __syncthreads();
        build_feat<C>(feat, xb, idxb, base, t, lane);
        __syncthreads();
        for (int ft = 0; ft < F / 16; ++ft) {
            const float* wrow = w + (ft * 16 + lm) * TC;
            v8f acc = {};
#pragma unroll
            for (int cc = 0; cc < TC / 4; ++cc) {
                v2f a  = *(const v2f*)&feat[lm][4 * cc + 2 * hi];
                v2f bm = *(const v2f*)&wrow[4 * cc + 2 * hi];
                acc = __builtin_amdgcn_wmma_f32_16x16x4_f32(false, a, false, bm,
                                                            (short)0, acc, false, false);
            }
            float s = 0.f, q = 0.f;
#pragma unroll
            for (int v = 0; v < 8; ++v) { s += acc[v]; q += acc[v] * acc[v]; }
            atomicAdd(&lsum[ft * 16 + lm], s);
            atomicAdd(&lsq[ft * 16 + lm], q);
        }
    }
    __syncthreads();
    for (int e = lane; e < F; e += 32) {
        atomicAdd(&gsum[e], lsum[e]);
        atomicAdd(&gsq[e], lsq[e]);
    }
}

// ---------------------------------------------------------------------------
// BN scale/bias per channel from accumulated moments.
// ---------------------------------------------------------------------------
__global__ void finalize_kernel(const float* __restrict__ gsum, const float* __restrict__ gsq,
                                const float* __restrict__ g, const float* __restrict__ bb,
                                float invcnt, int F,
                                float* __restrict__ scale, float* __restrict__ bias) {
    int f = blockIdx.x * blockDim.x + threadIdx.x;
    if (f >= F) return;
    float m   = gsum[f] * invcnt;
    float var = gsq[f] * invcnt - m * m;
    float sc  = g[f] * rsqrtf(var + 1e-5f);
    scale[f] = sc;
    bias[f]  = bb[f] - m * sc;
}

// ---------------------------------------------------------------------------
// EdgeConv pass 2: recompute GEMM, apply BN affine + leaky, max over k=20,
// write layer output into its channel slice of cat[B][512][N].
// ---------------------------------------------------------------------------
template <int C, int F>
__global__ __launch_bounds__(32) void edge_apply_kernel(
    const float* __restrict__ xin, int bstride,
    const float* __restrict__ w, const int* __restrict__ idx,
    const float* __restrict__ scale, const float* __restrict__ bias,
    float* __restrict__ cat, int chan_off) {
    constexpr int TC = 2 * C;
    constexpr int FP = TC + 2;
    __shared__ float feat[16][FP];
    __shared__ float hbuf[16][16];
    __shared__ float outbuf[4][F];

    int lane = threadIdx.x;
    int hi = lane >> 4, lm = lane & 15;
    int wid = blockIdx.x;
    int b = wid >> 10, ngrp = wid & 1023;
    int base = ngrp * 80;

    for (int e = lane; e < 4 * F; e += 32) outbuf[e / F][e - (e / F) * F] = -3.0e38f;
    const float* xb   = xin + (long)b * bstride;
    const int*   idxb = idx + (long)b * NPTS * KNN;

    for (int t = 0; t < 5; ++t) {
        __syncthreads();
        build_feat<C>(feat, xb, idxb, base, t, lane);
        __syncthreads();
        for (int ft = 0; ft < F / 16; ++ft) {
            const float* wrow = w + (ft * 16 + lm) * TC;
            v8f acc = {};
#pragma unroll
            for (int cc = 0; cc < TC / 4; ++cc) {
                v2f a  = *(const v2f*)&feat[lm][4 * cc + 2 * hi];
                v2f bm = *(const v2f*)&wrow[4 * cc + 2 * hi];
                acc = __builtin_amdgcn_wmma_f32_16x16x4_f32(false, a, false, bm,
                                                            (short)0, acc, false, false);
            }
            float sc = scale[ft * 16 + lm], bi = bias[ft * 16 + lm];
#pragma unroll
            for (int v = 0; v < 8; ++v) {
                float h = acc[v] * sc + bi;
                h = h > 0.f ? h : 0.2f * h;
                hbuf[v + 8 * hi][lm] = h;
            }
            __syncthreads();
            if (lane < 16) {
                for (int M = 0; M < 16; ++M) {
                    int mf = base + t * 16 + M;
                    int nl = mf / KNN - ngrp * 4;
                    float h = hbuf[M][lane];
                    if (h > outbuf[nl][ft * 16 + lane]) outbuf[nl][ft * 16 + lane] = h;
                }
            }
            __syncthreads();
        }
    }
    for (int e = lane; e < 4 * F; e += 32) {
        int nl = e / F, f = e - nl * F;
        cat[((long)b * 512 + chan_off + f) * NPTS + ngrp * 4 + nl] = outbuf[nl][f];
    }
}

// ---------------------------------------------------------------------------
// Final 1x1 conv: (B*N x 512) * (512 x 256) via WMMA; write pre-activations to
// d_out and accumulate BN moments per output channel.
// ---------------------------------------------------------------------------
__global__ __launch_bounds__(32) void final_stats_kernel(
    const float* __restrict__ cat, const float* __restrict__ wf,
    float* __restrict__ out, float* __restrict__ gsum, float* __restrict__ gsq) {
    __shared__ float lsum[256], lsq[256];
    int lane = threadIdx.x;
    int hi = lane >> 4, lm = lane & 15;
    int mt = blockIdx.x;             // b*(N/16) + nt
    int b = mt >> 8, nt = mt & 255;
    int n0 = nt * 16;

    for (int e = lane; e < 256; e += 32) { lsum[e] = 0.f; lsq[e] = 0.f; }
    const float* cb = cat + (long)b * 512 * NPTS;

    for (int ft = 0; ft < 16; ++ft) {
        const float* wrow = wf + (ft * 16 + lm) * 512;
        v8f acc = {};
        for (int c = 0; c < 512; c += 4) {
            v2f a, bm;
            a.x = cb[(c + 2 * hi) * NPTS + n0 + lm];
            a.y = cb[(c + 2 * hi + 1) * NPTS + n0 + lm];
            bm  = *(const v2f*)&wrow[c + 2 * hi];
            acc = __builtin_amdgcn_wmma_f32_16x16x4_f32(false, a, false, bm,
                                                        (short)0, acc, false, false);
        }
        float s = 0.f, q = 0.f;
#pragma unroll
        for (int v = 0; v < 8; ++v) {
            s += acc[v]; q += acc[v] * acc[v];
            out[((long)b * 256 + ft * 16 + lm) * NPTS + n0 + v + 8 * hi] = acc[v];
        }
        atomicAdd(&lsum[ft * 16 + lm], s);
        atomicAdd(&lsq[ft * 16 + lm], q);
    }
    __syncthreads();
    for (int e = lane; e < 256; e += 32) {
        atomicAdd(&gsum[e], lsum[e]);
        atomicAdd(&gsq[e], lsq[e]);
    }
}

__global__ void final_apply_kernel(float* __restrict__ out,
                                   const float* __restrict__ scale,
                                   const float* __restrict__ bias) {
    int i = blockIdx.x * blockDim.x + threadIdx.x;
    if (i >= NB * 256 * NPTS) return;
    int f = (i / NPTS) & 255;
    float v = out[i] * scale[f] + bias[f];
    out[i] = v > 0.f ? v : 0.2f * v;
}

// ---------------------------------------------------------------------------
template <int C, int F>
static void run_layer(const float* in, int bstr, const float* w, const float* g,
                      const float* bb, float* xx, int* idx, float* gsum, float* gsq,
                      float* scale, float* bias, float* cat, int chan_off,
                      hipStream_t stream) {
    xx_kernel<<<64, 256, 0, stream>>>(in, bstr, C, xx);
    knn_kernel<C><<<1024, 128, 0, stream>>>(in, bstr, xx, idx);
    zero_kernel<<<2, 256, 0, stream>>>(gsum, 512);
    edge_stats_kernel<C, F><<<4096, 32, 0, stream>>>(in, bstr, w, idx, gsum, gsq);
    finalize_kernel<<<1, 256, 0, stream>>>(gsum, gsq, g, bb, 1.0f / 327680.0f, F,
                                           scale, bias);
    edge_apply_kernel<C, F><<<4096, 32, 0, stream>>>(in, bstr, w, idx, scale, bias,
                                                     cat, chan_off);
}

extern "C" void kernel_launch(void* const* d_in, const int* in_sizes, int n_in,
                              void* d_out, int out_size, void* d_ws, size_t ws_size,
                              hipStream_t stream) {
    (void)in_sizes; (void)n_in; (void)out_size; (void)ws_size;
    const float* x  = (const float*)d_in[0];
    const float* wf = (const float*)d_in[14];
    const float* gf = (const float*)d_in[15];
    const float* bf = (const float*)d_in[16];
    float* out = (float*)d_out;

    // workspace layout (floats)
    float* ws    = (float*)d_ws;
    float* xx    = ws;                         // 4*4096
    int*   idx   = (int*)(ws + 16384);         // 4*4096*20
    float* cat   = ws + 16384 + 327680;        // 4*512*4096
    float* gsum  = cat + 8388608;              // 256
    float* gsq   = gsum + 256;                 // 256
    float* scale = gsq + 256;                  // 256
    float* bias  = scale + 256;                // 256

    // layer 0: in = x, C=128, F=64
    run_layer<128, 64>(x, 128 * NPTS, (const float*)d_in[2], (const float*)d_in[3],
                       (const float*)d_in[4], xx, idx, gsum, gsq, scale, bias, cat, 0,
                       stream);
    // layer 1: in = cat[:,0:64,:], C=64, F=64
    run_layer<64, 64>(cat + 0 * NPTS, 512 * NPTS, (const float*)d_in[5],
                      (const float*)d_in[6], (const float*)d_in[7], xx, idx, gsum, gsq,
                      scale, bias, cat, 64, stream);
    // layer 2: in = cat[:,64:128,:], C=64, F=128
    run_layer<64, 128>(cat + 64 * NPTS, 512 * NPTS, (const float*)d_in[8],
                       (const float*)d_in[9], (const float*)d_in[10], xx, idx, gsum, gsq,
                       scale, bias, cat, 128, stream);
    // layer 3: in = cat[:,128:256,:], C=128, F=256
    run_layer<128, 256>(cat + 128 * NPTS, 512 * NPTS, (const float*)d_in[11],
                        (const float*)d_in[12], (const float*)d_in[13], xx, idx, gsum,
                        gsq, scale, bias, cat, 256, stream);

    zero_kernel<<<2, 256, 0, stream>>>(gsum, 512);
    final_stats_kernel<<<1024, 32, 0, stream>>>(cat, wf, out, gsum, gsq);
    finalize_kernel<<<1, 256, 0, stream>>>(gsum, gsq, gf, bf, 1.0f / 16384.0f, 256,
                                           scale, bias);
    final_apply_kernel<<<16384, 256, 0, stream>>>(out, scale, bias);
}